// AttentionAggregator_33930241638752
// MI455X (gfx1250) — compile-verified
//
#include <hip/hip_runtime.h>
#include <math.h>

#define N_SRC   20000
#define N_DST   20000
#define NEDGE   640000
#define IN_DIM  256
#define OUT_DIM 128

typedef __attribute__((ext_vector_type(2))) float v2f;
typedef __attribute__((ext_vector_type(8))) float v8f;

// ---- order-preserving float<->uint encoding for integer atomicMax ----
__device__ __forceinline__ unsigned enc_f32(float x) {
  unsigned u = __float_as_uint(x);
  return (u & 0x80000000u) ? ~u : (u | 0x80000000u);
}
__device__ __forceinline__ float dec_f32(unsigned u) {
  return __uint_as_float((u & 0x80000000u) ? (u & 0x7fffffffu) : ~u);
}

// ---------------------------------------------------------------------
// Kernel 0: zero output + reset softmax accumulators
// ---------------------------------------------------------------------
__global__ __launch_bounds__(256) void init_kernel(float* __restrict__ out,
                                                   unsigned* __restrict__ gmaxBits,
                                                   float* __restrict__ gsum) {
  size_t i = (size_t)blockIdx.x * blockDim.x + threadIdx.x;
  size_t n = (size_t)N_DST * OUT_DIM;
  size_t stride = (size_t)gridDim.x * blockDim.x;
  for (; i < n; i += stride) out[i] = 0.0f;
  if (blockIdx.x == 0 && threadIdx.x == 0) {
    *gmaxBits = 0u;   // below any encoded finite score
    *gsum = 0.0f;
  }
}

// ---------------------------------------------------------------------
// Kernel 1: Y[M,128] = X[M,256] @ W[256,128] + b  (f32 WMMA 16x16x4)
// Block = 256 threads = 8 waves; wave w owns N-tile [w*16, w*16+16);
// block b owns M-tile [b*16, b*16+16).  M=20000 divisible by 16.
// ---------------------------------------------------------------------
__global__ __launch_bounds__(256) void proj_gemm(const float* __restrict__ X,
                                                 const float* __restrict__ W,
                                                 const float* __restrict__ bias,
                                                 float* __restrict__ Y) {
  const int wave = threadIdx.x >> 5;
  const int lane = threadIdx.x & 31;
  const int hi   = lane >> 4;    // 0: K pair {0,1}, 1: K pair {2,3}
  const int l    = lane & 15;
  const int m0   = blockIdx.x * 16;
  const int n0   = wave * 16;
  const int arow = m0 + l;       // A-matrix row for this lane
  const int bcol = n0 + l;       // B/C/D column for this lane

  const float* __restrict__ xrow = X + (size_t)arow * IN_DIM;

  v8f acc = {};
  #pragma unroll 4
  for (int k = 0; k < IN_DIM; k += 4) {
    const int ka = k + 2 * hi;
    v2f a, b;
    a.x = xrow[ka];
    a.y = xrow[ka + 1];
    b.x = W[(size_t)ka * OUT_DIM + bcol];
    b.y = W[(size_t)(ka + 1) * OUT_DIM + bcol];
    acc = __builtin_amdgcn_wmma_f32_16x16x4_f32(
        /*neg_a=*/false, a, /*neg_b=*/false, b,
        /*c_mod=*/(short)0, acc, /*reuse_a=*/false, /*reuse_b=*/false);
  }

  const float bv = bias[bcol];
  #pragma unroll
  for (int i = 0; i < 8; ++i) {
    const int row = m0 + i + 8 * hi;   // C/D layout: lanes16-31 hold M=i+8
    Y[(size_t)row * OUT_DIM + bcol] = acc[i] + bv;
  }
}

// ---------------------------------------------------------------------
// Kernel 2: per-edge dot-product scores + global max (one wave per edge)
// ---------------------------------------------------------------------
__global__ __launch_bounds__(256) void edge_scores(const float* __restrict__ srcProj,
                                                   const float* __restrict__ dstProj,
                                                   const int* __restrict__ e0,
                                                   const int* __restrict__ e1,
                                                   float* __restrict__ scores,
                                                   unsigned* __restrict__ gmaxBits) {
  const int edge = (int)(((size_t)blockIdx.x * blockDim.x + threadIdx.x) >> 5);
  const int lane = threadIdx.x & 31;
  if (edge >= NEDGE) return;

  const int s = e0[edge];
  const int d = e1[edge];
  const float4 a = ((const float4*)(srcProj + (size_t)s * OUT_DIM))[lane];
  const float4 b = ((const float4*)(dstProj + (size_t)d * OUT_DIM))[lane];
  float dot = a.x * b.x + a.y * b.y + a.z * b.z + a.w * b.w;

  #pragma unroll
  for (int off = 16; off >= 1; off >>= 1)
    dot += __shfl_xor(dot, off, 32);

  if (lane == 0) {
    scores[edge] = dot;
    atomicMax(gmaxBits, enc_f32(dot));
  }
}

// ---------------------------------------------------------------------
// Kernel 3: exp(score - max) in place + global sum (one thread per edge)
// ---------------------------------------------------------------------
__global__ __launch_bounds__(256) void edge_exp(float* __restrict__ scores,
                                                const unsigned* __restrict__ gmaxBits,
                                                float* __restrict__ gsum) {
  const int gid = (int)((size_t)blockIdx.x * blockDim.x + threadIdx.x);
  float v = 0.0f;
  if (gid < NEDGE) {
    const float m = dec_f32(*gmaxBits);
    v = __expf(scores[gid] - m);
    scores[gid] = v;
  }
  #pragma unroll
  for (int off = 16; off >= 1; off >>= 1)
    v += __shfl_xor(v, off, 32);
  if ((threadIdx.x & 31) == 0 && v != 0.0f)
    unsafeAtomicAdd(gsum, v);
}

// ---------------------------------------------------------------------
// Kernel 4: weighted scatter-add of messages (one wave per edge)
// ---------------------------------------------------------------------
__global__ __launch_bounds__(256) void edge_scatter(const float* __restrict__ srcProj,
                                                    const float* __restrict__ expScores,
                                                    const float* __restrict__ gsum,
                                                    const int* __restrict__ e0,
                                                    const int* __restrict__ e1,
                                                    float* __restrict__ out) {
  const int edge = (int)(((size_t)blockIdx.x * blockDim.x + threadIdx.x) >> 5);
  const int lane = threadIdx.x & 31;
  if (edge >= NEDGE) return;

  const float w = expScores[edge] / *gsum;
  const int s = e0[edge];
  const int d = e1[edge];
  const float4 a = ((const float4*)(srcProj + (size_t)s * OUT_DIM))[lane];
  float* o = out + (size_t)d * OUT_DIM + lane * 4;
  unsafeAtomicAdd(o + 0, a.x * w);
  unsafeAtomicAdd(o + 1, a.y * w);
  unsafeAtomicAdd(o + 2, a.z * w);
  unsafeAtomicAdd(o + 3, a.w * w);
}

// ---------------------------------------------------------------------
extern "C" void kernel_launch(void* const* d_in, const int* in_sizes, int n_in,
                              void* d_out, int out_size, void* d_ws, size_t ws_size,
                              hipStream_t stream) {
  const float* src = (const float*)d_in[0];
  const float* dst = (const float*)d_in[1];
  const int*   ei  = (const int*)d_in[2];          // [2, E] int32
  const float* Ws  = (const float*)d_in[3];
  const float* bs  = (const float*)d_in[4];
  const float* Wd  = (const float*)d_in[5];
  const float* bd  = (const float*)d_in[6];
  float* out = (float*)d_out;

  const int* e0 = ei;
  const int* e1 = ei + NEDGE;

  // workspace layout (all 16B-aligned)
  char* ws = (char*)d_ws;
  float*    srcProj  = (float*)(ws);
  float*    dstProj  = (float*)(ws + (size_t)N_SRC * OUT_DIM * 4);
  float*    scores   = (float*)(ws + (size_t)(N_SRC + N_DST) * OUT_DIM * 4);
  unsigned* gmaxBits = (unsigned*)(ws + (size_t)(N_SRC + N_DST) * OUT_DIM * 4 + (size_t)NEDGE * 4);
  float*    gsum     = (float*)(gmaxBits + 1);

  // 0: init output + accumulators
  init_kernel<<<(N_DST * OUT_DIM + 255) / 256, 256, 0, stream>>>(out, gmaxBits, gsum);

  // 1: projections (WMMA f32 16x16x4)
  proj_gemm<<<N_SRC / 16, 256, 0, stream>>>(src, Ws, bs, srcProj);
  proj_gemm<<<N_DST / 16, 256, 0, stream>>>(dst, Wd, bd, dstProj);

  // 2: edge scores + global max (one wave per edge: 8 edges / 256-thread block)
  edge_scores<<<NEDGE / 8, 256, 0, stream>>>(srcProj, dstProj, e0, e1, scores, gmaxBits);

  // 3: exp + global sum (one thread per edge)
  edge_exp<<<(NEDGE + 255) / 256, 256, 0, stream>>>(scores, gmaxBits, gsum);

  // 4: weighted scatter-add (one wave per edge)
  edge_scatter<<<NEDGE / 8, 256, 0, stream>>>(srcProj, scores, gsum, e0, e1, out);
}